// TransformerEncoderLayer_71055938945150
// MI455X (gfx1250) — compile-verified
//
#include <hip/hip_runtime.h>
#include <hip/hip_bf16.h>

typedef float v2f __attribute__((ext_vector_type(2)));
typedef float v8f __attribute__((ext_vector_type(8)));

// D = A(16x4) * B(4x16) + C, f32, wave32. SRC layout per CDNA5 ISA 7.12.2.
__device__ __forceinline__ v8f wmma4(v2f a, v2f b, v8f c) {
  return __builtin_amdgcn_wmma_f32_16x16x4_f32(false, a, false, b, (short)0, c,
                                               false, false);
}
__device__ __forceinline__ v8f vzero8() {
  v8f z = {0.f, 0.f, 0.f, 0.f, 0.f, 0.f, 0.f, 0.f};
  return z;
}

// ---------------------------------------------------------------------------
// Generic GEMM: C[M,N] = act(A[M,K] @ Bw[N,K]^T + bias), A row-major lda=K,
// Bw row-major (weight layout), C row-major with ldc. Grid: (M/128, N/64),
// 256 threads = 8 waves; wave (wm,wn) owns a 32x32 quad of 16x16 WMMA tiles.
// Requires M%128==0, N%64==0, K%4==0.
// ---------------------------------------------------------------------------
__global__ __launch_bounds__(256) void gemm_kernel(
    const float* __restrict__ A, const float* __restrict__ Bw,
    const float* __restrict__ bias, float* __restrict__ C,
    int K, int ldc, int relu) {
  const int tid  = threadIdx.x;
  const int w    = tid >> 5;
  const int lane = tid & 31;
  const int r    = lane & 15;
  const int half = lane >> 4;
  const int m0 = blockIdx.x * 128 + (w >> 1) * 32;
  const int n0 = blockIdx.y * 64  + (w & 1) * 32;

  const float* a0p = A  + (size_t)(m0 + r) * K      + 2 * half;
  const float* a1p = A  + (size_t)(m0 + 16 + r) * K + 2 * half;
  const float* b0p = Bw + (size_t)(n0 + r) * K      + 2 * half;
  const float* b1p = Bw + (size_t)(n0 + 16 + r) * K + 2 * half;

  v8f acc00 = vzero8(), acc01 = vzero8(), acc10 = vzero8(), acc11 = vzero8();

  for (int k = 0; k < K; k += 4) {
    v2f a0 = *(const v2f*)(a0p + k);
    v2f a1 = *(const v2f*)(a1p + k);
    v2f b0 = *(const v2f*)(b0p + k);
    v2f b1 = *(const v2f*)(b1p + k);
    acc00 = wmma4(a0, b0, acc00);
    acc01 = wmma4(a0, b1, acc01);
    acc10 = wmma4(a1, b0, acc10);
    acc11 = wmma4(a1, b1, acc11);
  }

  const float bia0 = bias ? bias[n0 + r] : 0.f;
  const float bia1 = bias ? bias[n0 + 16 + r] : 0.f;
#pragma unroll
  for (int v = 0; v < 8; v++) {
    const int row0 = m0 + v + 8 * half;
    const int row1 = row0 + 16;
    const int col0 = n0 + r;
    const int col1 = col0 + 16;
    float x;
    x = acc00[v] + bia0; if (relu) x = fmaxf(x, 0.f); C[(size_t)row0 * ldc + col0] = x;
    x = acc01[v] + bia1; if (relu) x = fmaxf(x, 0.f); C[(size_t)row0 * ldc + col1] = x;
    x = acc10[v] + bia0; if (relu) x = fmaxf(x, 0.f); C[(size_t)row1 * ldc + col0] = x;
    x = acc11[v] + bia1; if (relu) x = fmaxf(x, 0.f); C[(size_t)row1 * ldc + col1] = x;
  }
}

// ---------------------------------------------------------------------------
// Flash attention, hd=128, B=2 batch interleaved rows (row = q*2 + b).
// Grid: (Lq/16, heads, 2), 128 threads = 4 waves. Per WG: 16 q-rows.
// Online softmax over 64-key tiles; S and P via WMMA; scores stay in LDS.
// ---------------------------------------------------------------------------
__global__ __launch_bounds__(128) void attn_kernel(
    const float* __restrict__ Q, int q_ld,
    const float* __restrict__ Km, int k_ld,
    const float* __restrict__ Vm, int v_ld,
    float* __restrict__ O, int o_ld,
    int Lk, float scale) {
  const int h  = blockIdx.y;
  const int b  = blockIdx.z;
  const int q0 = blockIdx.x * 16;
  const int tid  = threadIdx.x;
  const int w    = tid >> 5;
  const int lane = tid & 31;
  const int r    = lane & 15;
  const int half = lane >> 4;

  __shared__ __align__(16) float Qs[16][128];
  __shared__ __align__(16) float Ss[16][64];
  __shared__ float red[16][8];
  __shared__ float mrow[16], lrow[16], frow[16];

  for (int i = tid; i < 16 * 128; i += 128) {
    const int qr = i >> 7, d = i & 127;
    Qs[qr][d] = Q[(size_t)((q0 + qr) * 2 + b) * q_ld + h * 128 + d];
  }
  if (tid < 16) { mrow[tid] = -3.0e38f; lrow[tid] = 0.f; }
  v8f accA = vzero8(), accB = vzero8();
  __syncthreads();

  const int row8 = tid >> 3, c8 = tid & 7;
  const int dA = w * 16, dB = (w + 4) * 16;

  for (int kt = 0; kt < Lk; kt += 64) {
    // --- S tile: wave w computes key columns [kt + w*16, +16) over hd=128
    {
      v8f s = vzero8();
      const float* kp = Km + (size_t)((kt + w * 16 + r) * 2 + b) * k_ld + h * 128 + 2 * half;
      const float* qp = &Qs[r][2 * half];
      for (int d = 0; d < 128; d += 4) {
        v2f av = *(const v2f*)(qp + d);
        v2f bv = *(const v2f*)(kp + d);
        s = wmma4(av, bv, s);
      }
#pragma unroll
      for (int v = 0; v < 8; v++) Ss[v + 8 * half][w * 16 + r] = s[v] * scale;
    }
    __syncthreads();
    // --- online softmax stats (8 lanes per q-row)
    float lm = -3.0e38f;
#pragma unroll
    for (int j = 0; j < 8; j++) lm = fmaxf(lm, Ss[row8][c8 * 8 + j]);
    red[row8][c8] = lm;
    __syncthreads();
    if (tid < 16) {
      float tm = red[tid][0];
      for (int j = 1; j < 8; j++) tm = fmaxf(tm, red[tid][j]);
      const float nm = fmaxf(mrow[tid], tm);
      frow[tid] = __expf(mrow[tid] - nm);
      mrow[tid] = nm;
    }
    __syncthreads();
    const float nm = mrow[row8];
    float ps = 0.f;
#pragma unroll
    for (int j = 0; j < 8; j++) {
      const float e = __expf(Ss[row8][c8 * 8 + j] - nm);
      Ss[row8][c8 * 8 + j] = e;
      ps += e;
    }
    red[row8][c8] = ps;
    __syncthreads();
    if (tid < 16) {
      float s2 = 0.f;
      for (int j = 0; j < 8; j++) s2 += red[tid][j];
      lrow[tid] = lrow[tid] * frow[tid] + s2;
    }
    __syncthreads();
    // --- rescale running O and accumulate P(16x64) @ V(64x16) tiles
#pragma unroll
    for (int v = 0; v < 8; v++) {
      const float f = frow[v + 8 * half];
      accA[v] *= f;
      accB[v] *= f;
    }
    for (int kk = 0; kk < 64; kk += 4) {
      v2f av = *(const v2f*)(&Ss[r][kk + 2 * half]);
      const float* v0 = Vm + (size_t)((kt + kk + 2 * half) * 2 + b) * v_ld + h * 128;
      const float* v1 = Vm + (size_t)((kt + kk + 2 * half + 1) * 2 + b) * v_ld + h * 128;
      v2f bA; bA.x = v0[dA + r]; bA.y = v1[dA + r];
      v2f bB; bB.x = v0[dB + r]; bB.y = v1[dB + r];
      accA = wmma4(av, bA, accA);
      accB = wmma4(av, bB, accB);
    }
    __syncthreads();
  }

  if (tid < 16) frow[tid] = 1.f / lrow[tid];
  __syncthreads();
#pragma unroll
  for (int v = 0; v < 8; v++) {
    const int qr = v + 8 * half;
    const float inv = frow[qr];
    const size_t orow = (size_t)((q0 + qr) * 2 + b) * o_ld + h * 128;
    O[orow + dA + r] = accA[v] * inv;
    O[orow + dB + r] = accB[v] * inv;
  }
}

// ---------------------------------------------------------------------------
// LayerNorm. mode 0: out = ln(x+y); mode 1: out = ln(x) + y. One block/row.
// ---------------------------------------------------------------------------
__global__ __launch_bounds__(256) void ln_kernel(
    const float* __restrict__ X, const float* __restrict__ Y,
    const float* __restrict__ g, const float* __restrict__ be,
    float* __restrict__ Out, int cols, int mode) {
  const int row = blockIdx.x;
  const int tid = threadIdx.x;
  const float* x = X + (size_t)row * cols;
  const float* y = Y + (size_t)row * cols;
  __shared__ float vbuf[1024];
  __shared__ float rs[256], rq[256];
  float s = 0.f, q = 0.f;
  for (int i = tid; i < cols; i += 256) {
    const float v = (mode == 0) ? (x[i] + y[i]) : x[i];
    vbuf[i] = v;
    s += v;
    q += v * v;
  }
  rs[tid] = s; rq[tid] = q;
  __syncthreads();
  for (int off = 128; off > 0; off >>= 1) {
    if (tid < off) { rs[tid] += rs[tid + off]; rq[tid] += rq[tid + off]; }
    __syncthreads();
  }
  const float mean = rs[0] / (float)cols;
  const float var  = rq[0] / (float)cols - mean * mean;
  const float inv  = rsqrtf(var + 1e-5f);
  for (int i = tid; i < cols; i += 256) {
    float v = (vbuf[i] - mean) * inv * g[i] + be[i];
    if (mode == 1) v += y[i];
    Out[(size_t)row * cols + i] = v;
  }
}

// ---------------------------------------------------------------------------
// Feature-branch precompute: rank-1 collapse of pl1 + feature MHA + pl2/pl3.
// ---------------------------------------------------------------------------
// u[i] = fa_wi[i]·pl1_w ; c[i] = fa_wi[i]·pl1_b + fa_bi[i]   (i in 0..383)
__global__ void pk1_kernel(const float* __restrict__ fa_wi,
                           const float* __restrict__ fa_bi,
                           const float* __restrict__ w1,
                           const float* __restrict__ b1,
                           float* __restrict__ u, float* __restrict__ c) {
  const int i = blockIdx.x * 128 + threadIdx.x;
  float su = 0.f, sc = 0.f;
  for (int j = 0; j < 128; j++) {
    const float w = fa_wi[i * 128 + j];
    su += w * w1[j];
    sc += w * b1[j];
  }
  u[i] = su;
  c[i] = sc + fa_bi[i];
}

// coef[h]=alpha_h=uq_h·uk_h, coef[4+h]=gamma_h=cq_h·uk_h,
// g[h*128+i] = (Wo[:,head h] @ uv_h)[i],  ccv = Wo @ cv + fa_bo
__global__ void pk2_kernel(const float* __restrict__ fa_wo,
                           const float* __restrict__ fa_bo,
                           const float* __restrict__ u,
                           const float* __restrict__ c,
                           float* __restrict__ coef, float* __restrict__ g,
                           float* __restrict__ ccv) {
  const int i = threadIdx.x;  // 0..127
  for (int h = 0; h < 4; h++) {
    float s = 0.f;
    for (int d = 0; d < 32; d++)
      s += fa_wo[i * 128 + h * 32 + d] * u[256 + h * 32 + d];
    g[h * 128 + i] = s;
  }
  float s = 0.f;
  for (int d = 0; d < 128; d++) s += fa_wo[i * 128 + d] * c[256 + d];
  ccv[i] = s + fa_bo[i];
  if (i < 4) {
    float a = 0.f;
    for (int d = 0; d < 32; d++) a += u[i * 32 + d] * u[128 + i * 32 + d];
    coef[i] = a;
  } else if (i < 8) {
    const int h = i - 4;
    float gm = 0.f;
    for (int d = 0; d < 32; d++) gm += c[h * 32 + d] * u[128 + h * 32 + d];
    coef[i] = gm;
  }
}

// G2[j][h] = pl2_w[j]·g_h ; c2[j] = pl2_w[j]·ccv + pl2_b[j]
__global__ void pk3_kernel(const float* __restrict__ W2,
                           const float* __restrict__ b2,
                           const float* __restrict__ g,
                           const float* __restrict__ ccv,
                           float* __restrict__ G2, float* __restrict__ c2) {
  const int j = blockIdx.x * 128 + threadIdx.x;  // 0..2047
  float a0 = 0.f, a1 = 0.f, a2 = 0.f, a3 = 0.f, ac = 0.f;
  for (int i = 0; i < 128; i++) {
    const float w = W2[j * 128 + i];
    a0 += w * g[i];
    a1 += w * g[128 + i];
    a2 += w * g[256 + i];
    a3 += w * g[384 + i];
    ac += w * ccv[i];
  }
  G2[j * 4 + 0] = a0; G2[j * 4 + 1] = a1;
  G2[j * 4 + 2] = a2; G2[j * 4 + 3] = a3;
  c2[j] = ac + b2[j];
}

// Feature-branch main: per (f,n) compute head softmaxes over the 128 features
// of column n, then the fused relu(r·G2 + c2)·w3 reduction. 256 thr = 2 n x 128 f.
__global__ __launch_bounds__(256) void feat_kernel(
    const float* __restrict__ src, const float* __restrict__ coef,
    const float* __restrict__ G2, const float* __restrict__ c2,
    const float* __restrict__ w3, const float* __restrict__ b3,
    float* __restrict__ z) {
  const int tid  = threadIdx.x;
  const int nloc = tid >> 7;
  const int f    = tid & 127;
  const int n    = blockIdx.x * 2 + nloc;
  __shared__ float scol[2][128];
  __shared__ float stage[256 * 6];  // [0,1024): G2 chunk, [1024,1280): c2, [1280,1536): w3
  scol[nloc][f] = src[(size_t)n * 128 + f];
  __syncthreads();

  const float tf = scol[nloc][f];
  const float invs = 0.17677669529663687f;  // 1/sqrt(32)
  float r[4];
#pragma unroll
  for (int h = 0; h < 4; h++) {
    const float ch = (coef[h] * tf + coef[4 + h]) * invs;
    float mx = -3.0e38f;
    for (int k = 0; k < 128; k++) mx = fmaxf(mx, ch * scol[nloc][k]);
    float se = 0.f, st = 0.f;
    for (int k = 0; k < 128; k++) {
      const float t = scol[nloc][k];
      const float e = __expf(ch * t - mx);
      se += e;
      st += e * t;
    }
    r[h] = st / se;
  }

  float acc = 0.f;
  for (int base = 0; base < 2048; base += 256) {
    __syncthreads();
    {
      const int j = base + tid;
      stage[tid * 4 + 0] = G2[j * 4 + 0];
      stage[tid * 4 + 1] = G2[j * 4 + 1];
      stage[tid * 4 + 2] = G2[j * 4 + 2];
      stage[tid * 4 + 3] = G2[j * 4 + 3];
      stage[1024 + tid]  = c2[j];
      stage[1280 + tid]  = w3[j];
    }
    __syncthreads();
    for (int jj = 0; jj < 256; jj++) {
      const float* gr = &stage[jj * 4];
      float v = stage[1024 + jj] + r[0] * gr[0] + r[1] * gr[1] +
                r[2] * gr[2] + r[3] * gr[3];
      v = fmaxf(v, 0.f);
      acc += v * stage[1280 + jj];
    }
  }
  z[(size_t)n * 128 + f] = acc + b3[0];
}

// ---------------------------------------------------------------------------
// Host driver
// ---------------------------------------------------------------------------
extern "C" void kernel_launch(void* const* d_in, const int* in_sizes, int n_in,
                              void* d_out, int out_size, void* d_ws, size_t ws_size,
                              hipStream_t stream) {
  const float* src   = (const float*)d_in[0];
  const float* sa_wi = (const float*)d_in[2];
  const float* sa_bi = (const float*)d_in[3];
  const float* sa_wo = (const float*)d_in[4];
  const float* sa_bo = (const float*)d_in[5];
  const float* fa_wi = (const float*)d_in[6];
  const float* fa_bi = (const float*)d_in[7];
  const float* fa_wo = (const float*)d_in[8];
  const float* fa_bo = (const float*)d_in[9];
  const float* pl1_w = (const float*)d_in[10];
  const float* pl1_b = (const float*)d_in[11];
  const float* pl2_w = (const float*)d_in[12];
  const float* pl2_b = (const float*)d_in[13];
  const float* pl3_w = (const float*)d_in[14];
  const float* pl3_b = (const float*)d_in[15];
  const float* pl4_w = (const float*)d_in[16];
  const float* pl4_b = (const float*)d_in[17];
  const float* pl5_w = (const float*)d_in[18];
  const float* pl5_b = (const float*)d_in[19];
  const float* pl6_w = (const float*)d_in[20];
  const float* pl6_b = (const float*)d_in[21];
  const float* pl7_w = (const float*)d_in[22];
  const float* pl7_b = (const float*)d_in[23];
  const float* l1_w  = (const float*)d_in[24];
  const float* l1_b  = (const float*)d_in[25];
  const float* l2_w  = (const float*)d_in[26];
  const float* l2_b  = (const float*)d_in[27];
  const float* pn1_g = (const float*)d_in[28];
  const float* pn1_b = (const float*)d_in[29];
  const float* pn2_g = (const float*)d_in[30];
  const float* pn2_b = (const float*)d_in[31];
  const float* n1_g  = (const float*)d_in[32];
  const float* n1_b  = (const float*)d_in[33];
  const float* n2_g  = (const float*)d_in[34];
  const float* n2_b  = (const float*)d_in[35];

  float* W = (float*)d_ws;
  size_t off = 0;
  auto take = [&](size_t n) { float* r = W + off; off += n; return r; };
  float* u_    = take(384);
  float* c_    = take(384);
  float* coef  = take(8);
  float* gvec  = take(512);
  float* ccvec = take(128);
  float* G2    = take(2048 * 4);
  float* c2    = take(2048);
  float* zbuf  = take(2304 * 128);
  float* src1  = take(2304 * 128);
  float* src1_ = take((size_t)2304 * 1024);
  float* Hbuf  = take((size_t)2304 * 2048);
  float* QKV   = take((size_t)2048 * 3072);
  float* O1    = take((size_t)2048 * 1024);
  float* T1    = take((size_t)2048 * 1024);
  float* SRC2  = take((size_t)2304 * 1024);
  float* SLu   = take((size_t)2048 * 1024);
  float* Qc    = take((size_t)256 * 1024);
  float* KVc   = take((size_t)2048 * 2048);
  float* SO    = take((size_t)2304 * 1024);
  float* FF    = take((size_t)2304 * 128);
  (void)ws_size; (void)in_sizes; (void)n_in; (void)out_size;

  const float scale = 0.08838834764831845f;  // 1/sqrt(128)

  // ---- Feature branch (rank-1 collapsed) ----
  pk1_kernel<<<3, 128, 0, stream>>>(fa_wi, fa_bi, pl1_w, pl1_b, u_, c_);
  pk2_kernel<<<1, 128, 0, stream>>>(fa_wo, fa_bo, u_, c_, coef, gvec, ccvec);
  pk3_kernel<<<16, 128, 0, stream>>>(pl2_w, pl2_b, gvec, ccvec, G2, c2);
  feat_kernel<<<1152, 256, 0, stream>>>(src, coef, G2, c2, pl3_w, pl3_b, zbuf);
  ln_kernel<<<2304, 256, 0, stream>>>(zbuf, src, pn1_g, pn1_b, src1, 128, 0);

  // ---- src1_ = relu(src1 @ pl4^T + b4) @ pl5^T + b5 ----
  gemm_kernel<<<dim3(18, 32), 256, 0, stream>>>(src1, pl4_w, pl4_b, Hbuf, 128, 2048, 1);
  gemm_kernel<<<dim3(18, 16), 256, 0, stream>>>(Hbuf, pl5_w, pl5_b, src1_, 2048, 1024, 0);

  // ---- self attention on li = rows [0,2048) of src1_ ----
  gemm_kernel<<<dim3(16, 48), 256, 0, stream>>>(src1_, sa_wi, sa_bi, QKV, 1024, 3072, 0);
  attn_kernel<<<dim3(64, 8, 2), 128, 0, stream>>>(QKV, 3072, QKV + 1024, 3072,
                                                  QKV + 2048, 3072, O1, 1024,
                                                  1024, scale);
  gemm_kernel<<<dim3(16, 16), 256, 0, stream>>>(O1, sa_wo, sa_bo, T1, 1024, 1024, 0);
  ln_kernel<<<2048, 256, 0, stream>>>(T1, src1_, pn2_g, pn2_b, SRC2, 1024, 0);  // sl

  // ---- sl_ = ln(relu(sl @ pl6^T) @ pl7^T, pn2) + sl ----
  gemm_kernel<<<dim3(16, 32), 256, 0, stream>>>(SRC2, pl6_w, pl6_b, Hbuf, 1024, 2048, 1);
  gemm_kernel<<<dim3(16, 16), 256, 0, stream>>>(Hbuf, pl7_w, pl7_b, T1, 2048, 1024, 0);
  ln_kernel<<<2048, 256, 0, stream>>>(T1, SRC2, pn2_g, pn2_b, SLu, 1024, 1);  // sl_

  // ---- cross attention: q = rows [2048,2304) of src1_, kv = sl_ ----
  gemm_kernel<<<dim3(2, 16), 256, 0, stream>>>(src1_ + (size_t)2048 * 1024, sa_wi,
                                               sa_bi, Qc, 1024, 1024, 0);
  gemm_kernel<<<dim3(16, 32), 256, 0, stream>>>(SLu, sa_wi + (size_t)1024 * 1024,
                                                sa_bi + 1024, KVc, 1024, 2048, 0);
  attn_kernel<<<dim3(8, 8, 2), 128, 0, stream>>>(Qc, 1024, KVc, 2048, KVc + 1024,
                                                 2048, O1, 1024, 1024, scale);
  gemm_kernel<<<dim3(2, 16), 256, 0, stream>>>(O1, sa_wo, sa_bo,
                                               SRC2 + (size_t)2048 * 1024, 1024,
                                               1024, 0);  // sr -> src2 tail

  // ---- so = ln(src2 + src1_); ff = relu(so @ l1^T) @ l2^T; out = ln(src1+ff) ----
  ln_kernel<<<2304, 256, 0, stream>>>(SRC2, src1_, n1_g, n1_b, SO, 1024, 0);
  gemm_kernel<<<dim3(18, 32), 256, 0, stream>>>(SO, l1_w, l1_b, Hbuf, 1024, 2048, 1);
  gemm_kernel<<<dim3(18, 2), 256, 0, stream>>>(Hbuf, l2_w, l2_b, FF, 2048, 128, 0);
  ln_kernel<<<2304, 256, 0, stream>>>(FF, src1, n2_g, n2_b, (float*)d_out, 128, 0);
}